// TGAR_60842506715527
// MI455X (gfx1250) — compile-verified
//
#include <hip/hip_runtime.h>
#include <math.h>

// Problem constants (match reference).
#define N_NODES 50000
#define N_EDGES 800000
#define E_SL    (N_EDGES + N_NODES)   // self-loops appended for GAT
#define HID     128
#define NF_TOT  ((size_t)N_NODES * HID)

typedef float v2f __attribute__((ext_vector_type(2)));
typedef float v8f __attribute__((ext_vector_type(8)));

// ---------------- helpers ----------------
__device__ __forceinline__ unsigned f2mono(float f) {
  unsigned u = __float_as_uint(f);
  return (u & 0x80000000u) ? ~u : (u | 0x80000000u);
}
__device__ __forceinline__ float mono2f(unsigned u) {
  return __uint_as_float((u & 0x80000000u) ? (u & 0x7fffffffu) : ~u);
}

// ---------------- WMMA GEMM: Out[n,dout] = act(A[n,din] @ W[din,dout] + b) ----
// One wave -> 16x64 tile via 4x V_WMMA_F32_16X16X4_F32 accumulators.
// act: 0 = none, 1 = relu
__global__ __launch_bounds__(256) void k_gemm(
    const float* __restrict__ A, const float* __restrict__ W,
    const float* __restrict__ bias, float* __restrict__ Out,
    int nrows, int din, int dout, int act)
{
  const int wave = threadIdx.x >> 5;
  const int lane = threadIdx.x & 31;
  const int r0 = (blockIdx.x * 8 + wave) * 16;
  if (r0 >= nrows) return;                       // wave-uniform exit
  const int c0   = blockIdx.y * 64;
  const int half = lane >> 4;                    // 0 or 1
  const int l15  = lane & 15;
  const int kOff = half * 2;
  const float* Ap  = A + (size_t)(r0 + l15) * din + kOff;
  const float* Bp0 = W + (size_t)kOff * dout + (c0 + l15);

  v8f acc0 = {}; v8f acc1 = {}; v8f acc2 = {}; v8f acc3 = {};
  for (int k = 0; k < din; k += 4) {
    v2f a; a.x = Ap[k]; a.y = Ap[k + 1];         // contiguous -> b64 load
    const float* Bp = Bp0 + (size_t)k * dout;
    v2f b0, b1, b2, b3;
    b0.x = Bp[0];   b0.y = Bp[dout];
    b1.x = Bp[16];  b1.y = Bp[dout + 16];
    b2.x = Bp[32];  b2.y = Bp[dout + 32];
    b3.x = Bp[48];  b3.y = Bp[dout + 48];
    acc0 = __builtin_amdgcn_wmma_f32_16x16x4_f32(false, a, false, b0, (short)0, acc0, false, false);
    acc1 = __builtin_amdgcn_wmma_f32_16x16x4_f32(false, a, false, b1, (short)0, acc1, false, false);
    acc2 = __builtin_amdgcn_wmma_f32_16x16x4_f32(false, a, false, b2, (short)0, acc2, false, false);
    acc3 = __builtin_amdgcn_wmma_f32_16x16x4_f32(false, a, false, b3, (short)0, acc3, false, false);
  }

#define STORE_TILE(ACC, TOFF)                                        \
  {                                                                  \
    int col = c0 + (TOFF) + l15;                                     \
    float bb = bias ? bias[col] : 0.f;                               \
    _Pragma("unroll")                                                \
    for (int i = 0; i < 8; ++i) {                                    \
      int row = r0 + i + half * 8;                                   \
      float v = ACC[i] + bb;                                         \
      if (act == 1) v = v > 0.f ? v : 0.f;                           \
      Out[(size_t)row * dout + col] = v;                             \
    }                                                                \
  }
  STORE_TILE(acc0, 0) STORE_TILE(acc1, 16) STORE_TILE(acc2, 32) STORE_TILE(acc3, 48)
#undef STORE_TILE
}

// ---------------- elementwise / init kernels ----------------
__global__ void k_zero(float* __restrict__ p, int n) {
  int t = blockIdx.x * blockDim.x + threadIdx.x;
  if (t < n) p[t] = 0.f;
}
__global__ void k_init_attn(unsigned* __restrict__ nmax, float* __restrict__ nsum, int n) {
  int t = blockIdx.x * blockDim.x + threadIdx.x;
  if (t < n) { nmax[t] = 0x007fffffu; /* encode(-inf) */ nsum[t] = 0.f; }
}
__global__ void k_mul(const float* __restrict__ a, const float* __restrict__ b,
                      float* __restrict__ o, int n) {
  int t = blockIdx.x * blockDim.x + threadIdx.x;
  if (t < n) o[t] = a[t] * b[t];
}
__global__ void k_add_relu(const float* __restrict__ a, const float* __restrict__ b,
                           float* __restrict__ o, int n) {
  int t = blockIdx.x * blockDim.x + threadIdx.x;
  if (t < n) { float v = a[t] + b[t]; o[t] = v > 0.f ? v : 0.f; }
}

// ---------------- GAT ----------------
// per-node attention coefficients: one block (128 thr) per node, wave w == head w
__global__ __launch_bounds__(128) void k_gat_coeff(
    const float* __restrict__ xh, const float* __restrict__ att_src,
    const float* __restrict__ att_dst, float* __restrict__ asrc,
    float* __restrict__ adst)
{
  int n = blockIdx.x;
  int t = threadIdx.x;                       // 0..127  -> head = t>>5
  float v = xh[(size_t)n * HID + t];
  float s = v * att_src[t];
  float d = v * att_dst[t];
  for (int off = 16; off; off >>= 1) {
    s += __shfl_xor(s, off, 32);
    d += __shfl_xor(d, off, 32);
  }
  if ((t & 31) == 0) {
    int h = t >> 5;
    asrc[n * 4 + h] = s;
    adst[n * 4 + h] = d;
  }
}

// pass 1: leaky-relu logits + segment max (thread per (edge, head))
__global__ void k_gat_logit(const int* __restrict__ src, const int* __restrict__ dst,
                            const float* __restrict__ asrc, const float* __restrict__ adst,
                            float* __restrict__ alpha, unsigned* __restrict__ nmax,
                            int E, int Etot)
{
  int t = blockIdx.x * blockDim.x + threadIdx.x;
  if (t >= Etot * 4) return;
  int e = t >> 2, h = t & 3;
  int s = e < E ? src[e] : (e - E);
  int d = e < E ? dst[e] : (e - E);
  float l = asrc[s * 4 + h] + adst[d * 4 + h];
  l = l > 0.f ? l : 0.2f * l;
  alpha[t] = l;
  atomicMax(&nmax[d * 4 + h], f2mono(l));
}

// pass 2 (shared GAT/transformer): exp(logit - max) + segment sum
__global__ void k_soft_exp(const int* __restrict__ dst, const unsigned* __restrict__ nmax,
                           float* __restrict__ alpha, float* __restrict__ nsum,
                           int E, int Etot)
{
  int t = blockIdx.x * blockDim.x + threadIdx.x;
  if (t >= Etot * 4) return;
  int e = t >> 2, h = t & 3;
  int d = e < E ? dst[e] : (e - E);
  float m  = mono2f(nmax[d * 4 + h]);
  float ev = expf(alpha[t] - m);
  alpha[t] = ev;
  atomicAdd(&nsum[d * 4 + h], ev);
}

// pass 3: wave per edge scatter-add, lane = feature within head chunk
__global__ void k_gat_agg(const int* __restrict__ src, const int* __restrict__ dst,
                          const float* __restrict__ xh, const float* __restrict__ alpha,
                          const float* __restrict__ nsum, float* __restrict__ agg,
                          int E, int Etot)
{
  int t = blockIdx.x * blockDim.x + threadIdx.x;
  if (t >= Etot * 32) return;
  int e = t >> 5, lane = t & 31;
  int s = e < E ? src[e] : (e - E);
  int d = e < E ? dst[e] : (e - E);
  const float* xr = xh + (size_t)s * HID;
  float* orow = agg + (size_t)d * HID;
#pragma unroll
  for (int i = 0; i < 4; ++i) {
    float w = alpha[e * 4 + i] / (nsum[d * 4 + i] + 1e-16f);
    int idx = i * 32 + lane;
    atomicAdd(&orow[idx], w * xr[idx]);
  }
}

__global__ void k_gat_final(const float* __restrict__ agg, const float* __restrict__ bias,
                            float* __restrict__ out, int n)
{
  int t = blockIdx.x * blockDim.x + threadIdx.x;
  if (t >= n) return;
  float v = agg[t] + bias[t & (HID - 1)];
  out[t] = v > 0.f ? v : 0.f;
}

// ---------------- TransformerConv ----------------
// logits: wave per edge, chunk i == head i, full 32-lane dot reduce
__global__ void k_trans_alpha(const int* __restrict__ src, const int* __restrict__ dst,
                              const float* __restrict__ q, const float* __restrict__ k,
                              const float* __restrict__ ea, const float* __restrict__ we,
                              float* __restrict__ alpha, unsigned* __restrict__ nmax, int E)
{
  int t = blockIdx.x * blockDim.x + threadIdx.x;
  if (t >= E * 32) return;
  int e = t >> 5, lane = t & 31;
  int s = src[e], d = dst[e];
  float eav = fmaxf(ea[e], 0.f);
  const float* qr = q + (size_t)d * HID;
  const float* kr = k + (size_t)s * HID;
#pragma unroll
  for (int i = 0; i < 4; ++i) {
    int idx = i * 32 + lane;
    float val = qr[idx] * (kr[idx] + eav * we[idx]);
    for (int off = 16; off; off >>= 1) val += __shfl_xor(val, off, 32);
    if (lane == 0) {
      float l = val * 0.17677669529663687f;  // 1/sqrt(32)
      alpha[e * 4 + i] = l;
      atomicMax(&nmax[d * 4 + i], f2mono(l));
    }
  }
}

__global__ void k_trans_agg(const int* __restrict__ src, const int* __restrict__ dst,
                            const float* __restrict__ v, const float* __restrict__ ea,
                            const float* __restrict__ we, const float* __restrict__ alpha,
                            const float* __restrict__ nsum, float* __restrict__ agg, int E)
{
  int t = blockIdx.x * blockDim.x + threadIdx.x;
  if (t >= E * 32) return;
  int e = t >> 5, lane = t & 31;
  int s = src[e], d = dst[e];
  float eav = fmaxf(ea[e], 0.f);
  const float* vr = v + (size_t)s * HID;
  float* orow = agg + (size_t)d * HID;
#pragma unroll
  for (int i = 0; i < 4; ++i) {
    float w = alpha[e * 4 + i] / (nsum[d * 4 + i] + 1e-16f);
    int idx = i * 32 + lane;
    atomicAdd(&orow[idx], w * (vr[idx] + eav * we[idx]));
  }
}

// softmax over 128 features, wave per node
__global__ void k_row_softmax128(float* __restrict__ x, int n)
{
  int t = blockIdx.x * blockDim.x + threadIdx.x;
  if (t >= n * 32) return;
  int nd = t >> 5, lane = t & 31;
  float* r = x + (size_t)nd * HID;
  float v0 = r[lane], v1 = r[lane + 32], v2 = r[lane + 64], v3 = r[lane + 96];
  float m = fmaxf(fmaxf(v0, v1), fmaxf(v2, v3));
  for (int off = 16; off; off >>= 1) m = fmaxf(m, __shfl_xor(m, off, 32));
  v0 = expf(v0 - m); v1 = expf(v1 - m); v2 = expf(v2 - m); v3 = expf(v3 - m);
  float s = v0 + v1 + v2 + v3;
  for (int off = 16; off; off >>= 1) s += __shfl_xor(s, off, 32);
  float inv = 1.f / s;
  r[lane] = v0 * inv; r[lane + 32] = v1 * inv; r[lane + 64] = v2 * inv; r[lane + 96] = v3 * inv;
}

// gate: sigmoid(dot(concat[z,x4,z-x4], gw) + gb), wave per node
__global__ void k_gate(const float* __restrict__ z, const float* __restrict__ x4,
                       const float* __restrict__ gw, const float* __restrict__ gb,
                       float* __restrict__ G, int n)
{
  int t = blockIdx.x * blockDim.x + threadIdx.x;
  if (t >= n * 32) return;
  int nd = t >> 5, lane = t & 31;
  const float* zr = z + (size_t)nd * HID;
  const float* xr = x4 + (size_t)nd * HID;
  float acc = 0.f;
#pragma unroll
  for (int i = 0; i < 4; ++i) {
    int idx = i * 32 + lane;
    float zv = zr[idx], xv = xr[idx];
    acc += zv * gw[idx] + xv * gw[HID + idx] + (zv - xv) * gw[2 * HID + idx];
  }
  for (int off = 16; off; off >>= 1) acc += __shfl_xor(acc, off, 32);
  if (lane == 0) G[nd] = 1.f / (1.f + expf(-(acc + gb[0])));
}

// out = leaky_relu(G*z + (1-G)*x4, 0.25)
__global__ void k_ctx_out(const float* __restrict__ z, const float* __restrict__ x4,
                          const float* __restrict__ G, float* __restrict__ out, int n)
{
  int t = blockIdx.x * blockDim.x + threadIdx.x;
  if (t >= n) return;
  float g = G[t >> 7];
  float v = g * z[t] + (1.f - g) * x4[t];
  out[t] = v > 0.f ? v : 0.25f * v;
}

// small head2 GEMM: [N,64]x[64,10]
__global__ void k_head2(const float* __restrict__ h, const float* __restrict__ w2,
                        const float* __restrict__ b2, float* __restrict__ lg)
{
  int t = blockIdx.x * blockDim.x + threadIdx.x;
  if (t >= N_NODES * 10) return;
  int n = t / 10, j = t - n * 10;
  const float* hr = h + (size_t)n * 64;
  float acc = b2[j];
#pragma unroll 8
  for (int k2 = 0; k2 < 64; ++k2) acc += hr[k2] * w2[k2 * 10 + j];
  lg[t] = acc;
}

__global__ void k_logsm10(const float* __restrict__ lg, float* __restrict__ out, int n)
{
  int t = blockIdx.x * blockDim.x + threadIdx.x;
  if (t >= n) return;
  const float* r = lg + (size_t)t * 10;
  float v[10]; float m = -INFINITY;
#pragma unroll
  for (int j = 0; j < 10; ++j) { v[j] = r[j]; m = fmaxf(m, v[j]); }
  float s = 0.f;
#pragma unroll
  for (int j = 0; j < 10; ++j) s += expf(v[j] - m);
  float ls = logf(s);
#pragma unroll
  for (int j = 0; j < 10; ++j) out[t * 10 + j] = v[j] - m - ls;
}

// ---------------- host-side orchestration ----------------
struct GatP { const float *att_dst, *att_src, *bias, *w; };
struct TrP  { const float *bk, *bq, *bskip, *bv, *we, *wk, *wq, *wskip, *wv; };

static inline int cdiv(long long a, int b) { return (int)((a + b - 1) / b); }

static void gemm(hipStream_t s, const float* A, const float* W, const float* b,
                 float* O, int n, int din, int dout, int act)
{
  dim3 g(cdiv((n + 15) / 16, 8), dout / 64);
  k_gemm<<<g, 256, 0, s>>>(A, W, b, O, n, din, dout, act);
}

static void run_gat(hipStream_t s, float* xio, const int* src, const int* dst,
                    const GatP& g, float* T1, float* T2, float* T3,
                    float* alpha, unsigned* nmax, float* nsum)
{
  float* asrc = T2;
  float* adst = T2 + (size_t)N_NODES * 4;
  gemm(s, xio, g.w, nullptr, T1, N_NODES, HID, HID, 0);                    // xh
  k_gat_coeff<<<N_NODES, 128, 0, s>>>(T1, g.att_src, g.att_dst, asrc, adst);
  k_init_attn<<<cdiv(N_NODES * 4, 256), 256, 0, s>>>(nmax, nsum, N_NODES * 4);
  k_zero<<<cdiv((long long)NF_TOT, 256), 256, 0, s>>>(T3, (int)NF_TOT);
  k_gat_logit<<<cdiv((long long)E_SL * 4, 256), 256, 0, s>>>(src, dst, asrc, adst,
                                                             alpha, nmax, N_EDGES, E_SL);
  k_soft_exp<<<cdiv((long long)E_SL * 4, 256), 256, 0, s>>>(dst, nmax, alpha, nsum,
                                                            N_EDGES, E_SL);
  k_gat_agg<<<cdiv((long long)E_SL * 32, 256), 256, 0, s>>>(src, dst, T1, alpha, nsum,
                                                            T3, N_EDGES, E_SL);
  k_gat_final<<<cdiv((long long)NF_TOT, 256), 256, 0, s>>>(T3, g.bias, xio, (int)NF_TOT);
}

static void run_ctx(hipStream_t s, const float* xin, const int* src, const int* dst,
                    const float* ea, const float* const* fcb, const float* const* fcw,
                    const float* gb, const float* gw, const GatP* g, const TrP& tp,
                    float* x1, float* x2, float* x3, float* x4,
                    float* T1, float* T2, float* T3,
                    float* alpha, unsigned* nmax, float* nsum, float* Gbuf, float* out)
{
  const int NFi = (int)NF_TOT;
  gemm(s, xin, fcw[0], fcb[0], x1, N_NODES, HID, HID, 1);
  gemm(s, xin, fcw[1], fcb[1], x2, N_NODES, HID, HID, 1);
  gemm(s, xin, fcw[2], fcb[2], x3, N_NODES, HID, HID, 1);
  gemm(s, xin, fcw[3], fcb[3], x4, N_NODES, HID, HID, 1);
  run_gat(s, x1, src, dst, g[0], T1, T2, T3, alpha, nmax, nsum);
  run_gat(s, x2, src, dst, g[1], T1, T2, T3, alpha, nmax, nsum);
  run_gat(s, x3, src, dst, g[2], T1, T2, T3, alpha, nmax, nsum);
  // transformer on xm = x1*x2
  k_mul<<<cdiv(NFi, 256), 256, 0, s>>>(x1, x2, T1, NFi);
  gemm(s, T1, tp.wq, tp.bq, x1, N_NODES, HID, HID, 0);        // q
  gemm(s, T1, tp.wk, tp.bk, x2, N_NODES, HID, HID, 0);        // k
  gemm(s, T1, tp.wv, tp.bv, T2, N_NODES, HID, HID, 0);        // v
  gemm(s, T1, tp.wskip, tp.bskip, T3, N_NODES, HID, HID, 0);  // skip
  k_init_attn<<<cdiv(N_NODES * 4, 256), 256, 0, s>>>(nmax, nsum, N_NODES * 4);
  k_trans_alpha<<<cdiv((long long)N_EDGES * 32, 256), 256, 0, s>>>(src, dst, x1, x2, ea,
                                                                   tp.we, alpha, nmax, N_EDGES);
  k_soft_exp<<<cdiv((long long)N_EDGES * 4, 256), 256, 0, s>>>(dst, nmax, alpha, nsum,
                                                               N_EDGES, N_EDGES);
  k_zero<<<cdiv(NFi, 256), 256, 0, s>>>(T1, NFi);
  k_trans_agg<<<cdiv((long long)N_EDGES * 32, 256), 256, 0, s>>>(src, dst, T2, ea, tp.we,
                                                                 alpha, nsum, T1, N_EDGES);
  k_add_relu<<<cdiv(NFi, 256), 256, 0, s>>>(T1, T3, T2, NFi);              // m
  k_row_softmax128<<<cdiv((long long)N_NODES * 32, 256), 256, 0, s>>>(T2, N_NODES);
  k_mul<<<cdiv(NFi, 256), 256, 0, s>>>(T2, x3, x3, NFi);                   // z
  k_gate<<<cdiv((long long)N_NODES * 32, 256), 256, 0, s>>>(x3, x4, gw, gb, Gbuf, N_NODES);
  k_ctx_out<<<cdiv(NFi, 256), 256, 0, s>>>(x3, x4, Gbuf, out, NFi);
}

extern "C" void kernel_launch(void* const* d_in, const int* in_sizes, int n_in,
                              void* d_out, int out_size, void* d_ws, size_t ws_size,
                              hipStream_t stream)
{
  (void)in_sizes; (void)n_in; (void)out_size; (void)ws_size;
  // JAX pytree (sorted-key) flattening of setup_inputs():
  // edge_attr, edge_index, params{fcs[10]{b,w}, fcsK{b,w}, gat[6]{att_dst,att_src,bias,w},
  //   head1{b,w}, head2{b,w}, trans[2]{bk,bq,bskip,bv,we,wk,wq,wskip,wv}}, x
  int p = 0;
  const float* edge_attr  = (const float*)d_in[p++];
  const int*   edge_index = (const int*)d_in[p++];
  const float *fcb[10], *fcw[10];
  for (int i = 0; i < 10; ++i) { fcb[i] = (const float*)d_in[p++]; fcw[i] = (const float*)d_in[p++]; }
  const float* fcsKb = (const float*)d_in[p++];
  const float* fcsKw = (const float*)d_in[p++];
  GatP gat[6];
  for (int i = 0; i < 6; ++i) {
    gat[i].att_dst = (const float*)d_in[p++];
    gat[i].att_src = (const float*)d_in[p++];
    gat[i].bias    = (const float*)d_in[p++];
    gat[i].w       = (const float*)d_in[p++];
  }
  const float* h1b = (const float*)d_in[p++];
  const float* h1w = (const float*)d_in[p++];
  const float* h2b = (const float*)d_in[p++];
  const float* h2w = (const float*)d_in[p++];
  TrP tr[2];
  for (int i = 0; i < 2; ++i) {
    tr[i].bk    = (const float*)d_in[p++];
    tr[i].bq    = (const float*)d_in[p++];
    tr[i].bskip = (const float*)d_in[p++];
    tr[i].bv    = (const float*)d_in[p++];
    tr[i].we    = (const float*)d_in[p++];
    tr[i].wk    = (const float*)d_in[p++];
    tr[i].wq    = (const float*)d_in[p++];
    tr[i].wskip = (const float*)d_in[p++];
    tr[i].wv    = (const float*)d_in[p++];
  }
  const float* x = (const float*)d_in[p++];

  const int* src = edge_index;
  const int* dst = edge_index + N_EDGES;

  // workspace carve (~246 MB of float)
  float* w = (float*)d_ws;
  float* x1 = w; w += NF_TOT;
  float* x2 = w; w += NF_TOT;
  float* x3 = w; w += NF_TOT;
  float* x4 = w; w += NF_TOT;
  float* T1 = w; w += NF_TOT;
  float* T2 = w; w += NF_TOT;
  float* T3 = w; w += NF_TOT;
  float* K1 = w; w += NF_TOT;
  float* Kf = w; w += NF_TOT;
  float* alpha = w; w += (size_t)E_SL * 4;
  unsigned* nmax = (unsigned*)w; w += (size_t)N_NODES * 4;
  float* nsum = w; w += (size_t)N_NODES * 4;
  float* Gbuf = w; w += N_NODES;

  run_ctx(stream, x, src, dst, edge_attr, &fcb[0], &fcw[0], fcb[4], fcw[4],
          &gat[0], tr[0], x1, x2, x3, x4, T1, T2, T3, alpha, nmax, nsum, Gbuf, K1);
  gemm(stream, K1, fcsKw, fcsKb, Kf, N_NODES, HID, HID, 0);
  run_ctx(stream, Kf, src, dst, edge_attr, &fcb[5], &fcw[5], fcb[9], fcw[9],
          &gat[3], tr[1], x1, x2, x3, x4, T1, T2, T3, alpha, nmax, nsum, Gbuf, K1);
  gemm(stream, K1, h1w, h1b, T1, N_NODES, HID, 64, 1);                 // h [N,64]
  k_head2<<<cdiv((long long)N_NODES * 10, 256), 256, 0, stream>>>(T1, h2w, h2b, T2);
  k_logsm10<<<cdiv(N_NODES, 256), 256, 0, stream>>>(T2, (float*)d_out, N_NODES);
}